// NTMCell_15049565405829
// MI455X (gfx1250) — compile-verified
//
#include <hip/hip_runtime.h>
#include <math.h>

// Problem constants
#define BB   64
#define NN   8192
#define DD   64
#define CC   256
#define INQ  128
#define CTRL 192
#define EPSF 1e-8f

typedef __attribute__((ext_vector_type(2))) float v2f;
typedef __attribute__((ext_vector_type(8))) float v8f;

__device__ __forceinline__ float softplusf(float x) {
    // stable: max(x,0) + log1p(exp(-|x|))
    return fmaxf(x, 0.f) + log1pf(expf(-fabsf(x)));
}
__device__ __forceinline__ float sigmoidf(float x) {
    return 1.f / (1.f + expf(-x));
}

// ---------------------------------------------------------------------------
// Block reductions (blockDim.x up to 1024, wave32)
// ---------------------------------------------------------------------------
__device__ __forceinline__ float blockReduceSum(float v, float* red) {
    __syncthreads();
#pragma unroll
    for (int off = 16; off > 0; off >>= 1) v += __shfl_xor(v, off, 32);
    const int wid = threadIdx.x >> 5, lane = threadIdx.x & 31;
    if (lane == 0) red[wid] = v;
    __syncthreads();
    if (wid == 0) {
        float r = (lane < (int)(blockDim.x >> 5)) ? red[lane] : 0.f;
#pragma unroll
        for (int off = 16; off > 0; off >>= 1) r += __shfl_xor(r, off, 32);
        if (lane == 0) red[0] = r;
    }
    __syncthreads();
    return red[0];
}

__device__ __forceinline__ float blockReduceMax(float v, float* red) {
    __syncthreads();
#pragma unroll
    for (int off = 16; off > 0; off >>= 1) v = fmaxf(v, __shfl_xor(v, off, 32));
    const int wid = threadIdx.x >> 5, lane = threadIdx.x & 31;
    if (lane == 0) red[wid] = v;
    __syncthreads();
    if (wid == 0) {
        float r = (lane < (int)(blockDim.x >> 5)) ? red[lane] : -INFINITY;
#pragma unroll
        for (int off = 16; off > 0; off >>= 1) r = fmaxf(r, __shfl_xor(r, off, 32));
        if (lane == 0) red[0] = r;
    }
    __syncthreads();
    return red[0];
}

// ---------------------------------------------------------------------------
// Kernel 1: controller  h = relu([x, prev_read_vec] @ W_ctrl + b_ctrl)
// M=64, K=192, N=256 using V_WMMA_F32_16X16X4_F32; one wave per 16x16 tile.
// K-loop split at the concat boundary (K=128) so each loop body is a pure
// branch-free load->wmma stream (no per-iteration cndmask/branches).
// A 16x4 layout: lanes 0-15 M=0..15 {K=kk,kk+1}; lanes 16-31 {K=kk+2,kk+3}.
// B 4x16 layout mirrors A; C/D: VGPR r -> M=r (lanes 0-15) / M=r+8 (16-31).
// ---------------------------------------------------------------------------
__global__ __launch_bounds__(32) void ntm_ctrl_wmma(
    const float* __restrict__ x, const float* __restrict__ prv,
    const float* __restrict__ Wc, const float* __restrict__ bc,
    float* __restrict__ h)
{
    const int n0 = blockIdx.x * 16;
    const int m0 = blockIdx.y * 16;
    const int lane = threadIdx.x;
    const int half = (lane < 16) ? 0 : 2;
    const int ml = lane & 15;
    const int m = m0 + ml;
    const int n = n0 + ml;
    v8f c = {};

    // K in [0,128): A sourced from x
    const float* xr = x + m * INQ;
#pragma unroll 4
    for (int kk = 0; kk < INQ; kk += 4) {
        const int ka = kk + half;
        v2f a, b;
        a.x = xr[ka];
        a.y = xr[ka + 1];
        b.x = Wc[ka * CC + n];
        b.y = Wc[(ka + 1) * CC + n];
        c = __builtin_amdgcn_wmma_f32_16x16x4_f32(false, a, false, b,
                                                  (short)0, c, false, false);
    }
    // K in [128,192): A sourced from prev_read_vector
    const float* pr = prv + m * DD;
#pragma unroll 4
    for (int kk = INQ; kk < CTRL; kk += 4) {
        const int ka = kk + half;
        v2f a, b;
        a.x = pr[ka - INQ];
        a.y = pr[ka + 1 - INQ];
        b.x = Wc[ka * CC + n];
        b.y = Wc[(ka + 1) * CC + n];
        c = __builtin_amdgcn_wmma_f32_16x16x4_f32(false, a, false, b,
                                                  (short)0, c, false, false);
    }

    const float bv = bc[n];
    const int rbase = m0 + ((lane < 16) ? 0 : 8);
#pragma unroll
    for (int r = 0; r < 8; ++r) {
        float v = c[r] + bv;
        h[(rbase + r) * CC + n] = v > 0.f ? v : 0.f;
    }
}

// ---------------------------------------------------------------------------
// Kernel 2: projections  out = act(h @ W + b), M=64, K=256, N=64 (WMMA f32)
// act: 0 = identity (keys), 1 = sigmoid (erase), 2 = tanh (add)
// ---------------------------------------------------------------------------
__global__ __launch_bounds__(32) void ntm_proj_wmma(
    const float* __restrict__ h, const float* __restrict__ W,
    const float* __restrict__ bias, float* __restrict__ out, int act)
{
    const int n0 = blockIdx.x * 16;
    const int m0 = blockIdx.y * 16;
    const int lane = threadIdx.x;
    const int half = (lane < 16) ? 0 : 2;
    const int ml = lane & 15;
    const int n = n0 + ml;
    const float* hr = h + (m0 + ml) * CC;
    v8f c = {};
#pragma unroll 4
    for (int kk = 0; kk < CC; kk += 4) {
        const int ka = kk + half;
        v2f a, b;
        a.x = hr[ka];
        a.y = hr[ka + 1];
        b.x = W[ka * DD + n];
        b.y = W[(ka + 1) * DD + n];
        c = __builtin_amdgcn_wmma_f32_16x16x4_f32(false, a, false, b,
                                                  (short)0, c, false, false);
    }
    const float bv = bias[n];
    const int rbase = m0 + ((lane < 16) ? 0 : 8);
#pragma unroll
    for (int r = 0; r < 8; ++r) {
        float v = c[r] + bv;
        if (act == 1) v = sigmoidf(v);
        else if (act == 2) v = tanhf(v);
        out[(rbase + r) * DD + n] = v;
    }
}

// ---------------------------------------------------------------------------
// Kernel 3: small scalar heads (beta, g, gamma, shift-softmax s) x {write,read}
// 8 jobs per batch element; 512 threads total.
// ---------------------------------------------------------------------------
__global__ void ntm_small(
    const float* __restrict__ h,
    const float* Wb_w, const float* bb_w, const float* Wg_w, const float* bg_w,
    const float* Wgam_w, const float* bgam_w, const float* Ws_w, const float* bs_w,
    const float* Wb_r, const float* bb_r, const float* Wg_r, const float* bg_r,
    const float* Wgam_r, const float* bgam_r, const float* Ws_r, const float* bs_r,
    float* beta_w, float* g_w, float* gam_w, float* s_w,
    float* beta_r, float* g_r, float* gam_r, float* s_r)
{
    const int t = blockIdx.x * blockDim.x + threadIdx.x;
    if (t >= BB * 8) return;
    const int b = t >> 3, j = t & 7;
    const float* hb = h + b * CC;
    const bool rd = (j >= 4);
    const int jj = j & 3;
    if (jj < 3) {
        const float* W; const float* bi;
        if (jj == 0)      { W = rd ? Wb_r   : Wb_w;   bi = rd ? bb_r   : bb_w;   }
        else if (jj == 1) { W = rd ? Wg_r   : Wg_w;   bi = rd ? bg_r   : bg_w;   }
        else              { W = rd ? Wgam_r : Wgam_w; bi = rd ? bgam_r : bgam_w; }
        float acc = bi[0];
        for (int i = 0; i < CC; ++i) acc += hb[i] * W[i];
        float v; float* o;
        if (jj == 0)      { v = softplusf(acc);       o = rd ? beta_r : beta_w; }
        else if (jj == 1) { v = sigmoidf(acc);        o = rd ? g_r    : g_w;    }
        else              { v = softplusf(acc) + 1.f; o = rd ? gam_r  : gam_w;  }
        o[b] = v;
    } else {
        const float* W  = rd ? Ws_r : Ws_w;
        const float* bi = rd ? bs_r : bs_w;
        float q0 = bi[0], q1 = bi[1], q2 = bi[2];
        for (int i = 0; i < CC; ++i) {
            const float hv = hb[i];
            q0 += hv * W[i * 3 + 0];
            q1 += hv * W[i * 3 + 1];
            q2 += hv * W[i * 3 + 2];
        }
        const float m  = fmaxf(q0, fmaxf(q1, q2));
        const float e0 = expf(q0 - m), e1 = expf(q1 - m), e2 = expf(q2 - m);
        const float iZ = 1.f / (e0 + e1 + e2);
        float* o = rd ? s_r : s_w;
        o[b * 3 + 0] = e0 * iZ; o[b * 3 + 1] = e1 * iZ; o[b * 3 + 2] = e2 * iZ;
    }
}

// ---------------------------------------------------------------------------
// Kernel 4: cosine similarity vs key over prev_memory. 16 lanes per row
// (row = 256 B = 2 cachelines, fully coalesced b128 loads).
// ---------------------------------------------------------------------------
__global__ __launch_bounds__(256) void ntm_sim(
    const float* __restrict__ mem, const float* __restrict__ key,
    float* __restrict__ sim)
{
    const int g = (blockIdx.x * blockDim.x + threadIdx.x) >> 4;
    const int lane = threadIdx.x & 15;
    if (g >= BB * NN) return;
    const int b = g >> 13;
    const float4 m4 = ((const float4*)(mem + (size_t)g * DD))[lane];
    const float4 k4 = ((const float4*)(key + b * DD))[lane];
    float dot = m4.x * k4.x + m4.y * k4.y + m4.z * k4.z + m4.w * k4.w;
    float msq = m4.x * m4.x + m4.y * m4.y + m4.z * m4.z + m4.w * m4.w;
    float ksq = k4.x * k4.x + k4.y * k4.y + k4.z * k4.z + k4.w * k4.w;
#pragma unroll
    for (int off = 8; off > 0; off >>= 1) {
        dot += __shfl_xor(dot, off, 32);
        msq += __shfl_xor(msq, off, 32);
        ksq += __shfl_xor(ksq, off, 32);
    }
    if (lane == 0) sim[g] = dot / (sqrtf(msq) * sqrtf(ksq) + EPSF);
}

// ---------------------------------------------------------------------------
// Kernel 5: recompute new_memory row (erase/add) + cosine sim vs read key.
// ---------------------------------------------------------------------------
__global__ __launch_bounds__(256) void ntm_write_sim(
    const float* __restrict__ mem, const float* __restrict__ w_w,
    const float* __restrict__ er, const float* __restrict__ ad,
    const float* __restrict__ key, float* __restrict__ sim)
{
    const int g = (blockIdx.x * blockDim.x + threadIdx.x) >> 4;
    const int lane = threadIdx.x & 15;
    if (g >= BB * NN) return;
    const int b = g >> 13;
    const float ww = w_w[g];
    const float4 m4 = ((const float4*)(mem + (size_t)g * DD))[lane];
    const float4 e4 = ((const float4*)(er + b * DD))[lane];
    const float4 a4 = ((const float4*)(ad + b * DD))[lane];
    float4 nm;
    nm.x = m4.x * (1.f - ww * e4.x) + ww * a4.x;
    nm.y = m4.y * (1.f - ww * e4.y) + ww * a4.y;
    nm.z = m4.z * (1.f - ww * e4.z) + ww * a4.z;
    nm.w = m4.w * (1.f - ww * e4.w) + ww * a4.w;
    const float4 k4 = ((const float4*)(key + b * DD))[lane];
    float dot = nm.x * k4.x + nm.y * k4.y + nm.z * k4.z + nm.w * k4.w;
    float msq = nm.x * nm.x + nm.y * nm.y + nm.z * nm.z + nm.w * nm.w;
    float ksq = k4.x * k4.x + k4.y * k4.y + k4.z * k4.z + k4.w * k4.w;
#pragma unroll
    for (int off = 8; off > 0; off >>= 1) {
        dot += __shfl_xor(dot, off, 32);
        msq += __shfl_xor(msq, off, 32);
        ksq += __shfl_xor(ksq, off, 32);
    }
    if (lane == 0) sim[g] = dot / (sqrtf(msq) * sqrtf(ksq) + EPSF);
}

// ---------------------------------------------------------------------------
// Kernel 6: addressing. In-place sim -> final weights. One block per batch.
// softmax(beta*sim) -> interpolate -> circular 3-tap shift -> pow(gamma) ->
// renormalize. wg kept in LDS for the shift.
// ---------------------------------------------------------------------------
__global__ __launch_bounds__(1024) void ntm_addr(
    float* __restrict__ sim, const float* __restrict__ prevw,
    const float* __restrict__ beta_, const float* __restrict__ g_,
    const float* __restrict__ gam_, const float* __restrict__ s_)
{
    __shared__ float wg[NN];
    __shared__ float red[32];
    const int b = blockIdx.x;
    const float beta = beta_[b], gv = g_[b], gamma = gam_[b];
    const float s0 = s_[b * 3 + 0], s1 = s_[b * 3 + 1], s2 = s_[b * 3 + 2];
    float* simb = sim + (size_t)b * NN;
    const float* pw = prevw + (size_t)b * NN;
    const int tid = threadIdx.x;

    float t[8];
    float mx = -INFINITY;
#pragma unroll
    for (int i = 0; i < 8; ++i) {
        t[i] = beta * simb[tid + i * 1024];
        mx = fmaxf(mx, t[i]);
    }
    mx = blockReduceMax(mx, red);

    float e[8]; float lsum = 0.f;
#pragma unroll
    for (int i = 0; i < 8; ++i) { e[i] = expf(t[i] - mx); lsum += e[i]; }
    const float invZ = 1.f / blockReduceSum(lsum, red);

#pragma unroll
    for (int i = 0; i < 8; ++i) {
        const int n = tid + i * 1024;
        wg[n] = gv * (e[i] * invZ) + (1.f - gv) * pw[n];
    }
    __syncthreads();

    float wp[8]; float psum = 0.f;
#pragma unroll
    for (int i = 0; i < 8; ++i) {
        const int n = tid + i * 1024;
        const float wsv = s0 * wg[(n + 1) & (NN - 1)] + s1 * wg[n] +
                          s2 * wg[(n - 1) & (NN - 1)];
        wp[i] = powf(wsv, gamma);
        psum += wp[i];
    }
    const float inv = 1.f / (blockReduceSum(psum, red) + EPSF);
#pragma unroll
    for (int i = 0; i < 8; ++i) simb[tid + i * 1024] = wp[i] * inv;
}

// ---------------------------------------------------------------------------
// Kernel 7: read_vec[b,:] = sum_n new_mem[b,n,:] * w_r[b,n] (new_mem
// recomputed on the fly), then emit output row [h(256) | read_vec(64)].
// One block (256 thr = 16 row-groups) per batch element.
// ---------------------------------------------------------------------------
__global__ __launch_bounds__(256) void ntm_read(
    const float* __restrict__ mem, const float* __restrict__ w_w,
    const float* __restrict__ er, const float* __restrict__ ad,
    const float* __restrict__ w_r, const float* __restrict__ h,
    float* __restrict__ out)
{
    __shared__ float acc[DD];
    const int b = blockIdx.x;
    if (threadIdx.x < DD) acc[threadIdx.x] = 0.f;
    __syncthreads();

    const int grp = threadIdx.x >> 4;
    const int lane = threadIdx.x & 15;
    const float4 e4 = ((const float4*)(er + b * DD))[lane];
    const float4 a4 = ((const float4*)(ad + b * DD))[lane];
    float4 s = {0.f, 0.f, 0.f, 0.f};
    for (int n = grp; n < NN; n += 16) {
        const size_t g = (size_t)b * NN + n;
        const float ww = w_w[g];
        const float wr = w_r[g];
        const float4 m4 = ((const float4*)(mem + g * DD))[lane];
        s.x += wr * (m4.x * (1.f - ww * e4.x) + ww * a4.x);
        s.y += wr * (m4.y * (1.f - ww * e4.y) + ww * a4.y);
        s.z += wr * (m4.z * (1.f - ww * e4.z) + ww * a4.z);
        s.w += wr * (m4.w * (1.f - ww * e4.w) + ww * a4.w);
    }
    atomicAdd(&acc[lane * 4 + 0], s.x);
    atomicAdd(&acc[lane * 4 + 1], s.y);
    atomicAdd(&acc[lane * 4 + 2], s.z);
    atomicAdd(&acc[lane * 4 + 3], s.w);
    __syncthreads();

    // output row: [h | read_vec]
    out[b * (CC + DD) + threadIdx.x] = h[b * CC + threadIdx.x];
    if (threadIdx.x < DD)
        out[b * (CC + DD) + CC + threadIdx.x] = acc[threadIdx.x];
}

// ---------------------------------------------------------------------------
// Launcher
// ---------------------------------------------------------------------------
extern "C" void kernel_launch(void* const* d_in, const int* in_sizes, int n_in,
                              void* d_out, int out_size, void* d_ws, size_t ws_size,
                              hipStream_t stream) {
    const float* x      = (const float*)d_in[0];
    const float* mem    = (const float*)d_in[1];
    const float* prw_r  = (const float*)d_in[2];
    const float* prw_w  = (const float*)d_in[3];
    const float* prv    = (const float*)d_in[4];
    const float* Wc     = (const float*)d_in[5];
    const float* bc     = (const float*)d_in[6];
    const float* Wk_r   = (const float*)d_in[7];
    const float* bk_r   = (const float*)d_in[8];
    const float* Wb_r   = (const float*)d_in[9];
    const float* bb_r   = (const float*)d_in[10];
    const float* Wg_r   = (const float*)d_in[11];
    const float* bg_r   = (const float*)d_in[12];
    const float* Ws_r   = (const float*)d_in[13];
    const float* bs_r   = (const float*)d_in[14];
    const float* Wgam_r = (const float*)d_in[15];
    const float* bgam_r = (const float*)d_in[16];
    const float* Wk_w   = (const float*)d_in[17];
    const float* bk_w   = (const float*)d_in[18];
    const float* Wb_w   = (const float*)d_in[19];
    const float* bb_w   = (const float*)d_in[20];
    const float* Wg_w   = (const float*)d_in[21];
    const float* bg_w   = (const float*)d_in[22];
    const float* Ws_w   = (const float*)d_in[23];
    const float* bs_w   = (const float*)d_in[24];
    const float* Wgam_w = (const float*)d_in[25];
    const float* bgam_w = (const float*)d_in[26];
    const float* We_w   = (const float*)d_in[27];
    const float* be_w   = (const float*)d_in[28];
    const float* Wa_w   = (const float*)d_in[29];
    const float* ba_w   = (const float*)d_in[30];

    float* ws = (float*)d_ws;
    float* h      = ws;            // 64*256
    float* kw     = h + BB * CC;   // 64*64
    float* kr     = kw + BB * DD;
    float* er     = kr + BB * DD;
    float* ad     = er + BB * DD;
    float* beta_w = ad + BB * DD;  // 64
    float* g_w    = beta_w + BB;
    float* gam_w  = g_w + BB;
    float* s_w    = gam_w + BB;    // 64*3
    float* beta_r = s_w + BB * 3;
    float* g_r    = beta_r + BB;
    float* gam_r  = g_r + BB;
    float* s_r    = gam_r + BB;    // 64*3
    float* simw   = s_r + BB * 3;  // 64*8192  (becomes w_w in place)
    float* simr   = simw + BB * NN; // 64*8192 (becomes w_r in place)

    // 1) controller (WMMA f32 16x16x4)
    ntm_ctrl_wmma<<<dim3(CC / 16, BB / 16), 32, 0, stream>>>(x, prv, Wc, bc, h);

    // 2) projections (WMMA)
    ntm_proj_wmma<<<dim3(DD / 16, BB / 16), 32, 0, stream>>>(h, Wk_w, bk_w, kw, 0);
    ntm_proj_wmma<<<dim3(DD / 16, BB / 16), 32, 0, stream>>>(h, Wk_r, bk_r, kr, 0);
    ntm_proj_wmma<<<dim3(DD / 16, BB / 16), 32, 0, stream>>>(h, We_w, be_w, er, 1);
    ntm_proj_wmma<<<dim3(DD / 16, BB / 16), 32, 0, stream>>>(h, Wa_w, ba_w, ad, 2);

    // 3) scalar heads
    ntm_small<<<2, 256, 0, stream>>>(h,
        Wb_w, bb_w, Wg_w, bg_w, Wgam_w, bgam_w, Ws_w, bs_w,
        Wb_r, bb_r, Wg_r, bg_r, Wgam_r, bgam_r, Ws_r, bs_r,
        beta_w, g_w, gam_w, s_w, beta_r, g_r, gam_r, s_r);

    const int rows = BB * NN;                 // 524288 rows
    const int blocksBig = rows / 16;          // 16 rows per 256-thr block

    // 4) write-addressing sims over prev_memory (134 MB read)
    ntm_sim<<<blocksBig, 256, 0, stream>>>(mem, kw, simw);

    // 5) write addressing -> w_w (in place)
    ntm_addr<<<BB, 1024, 0, stream>>>(simw, prw_w, beta_w, g_w, gam_w, s_w);

    // 6) recompute new_memory + read-key sims (134 MB read, L2-resident)
    ntm_write_sim<<<blocksBig, 256, 0, stream>>>(mem, simw, er, ad, kr, simr);

    // 7) read addressing -> w_r (in place)
    ntm_addr<<<BB, 1024, 0, stream>>>(simr, prw_r, beta_r, g_r, gam_r, s_r);

    // 8) read vector + final output assembly
    ntm_read<<<BB, 256, 0, stream>>>(mem, simw, er, ad, simr, h, (float*)d_out);
}